// GCNEncoder_54030688584367
// MI455X (gfx1250) — compile-verified
//
#include <hip/hip_runtime.h>

// Problem constants from the reference.
#define NN   50000   // nodes
#define NE   800000  // edges
#define INC  256
#define HID  128
#define LAT  64

typedef float v2f __attribute__((ext_vector_type(2)));
typedef float v8f __attribute__((ext_vector_type(8)));

// ---------------------------------------------------------------- degree ----
__global__ void k_init_deg(float* __restrict__ deg) {
    int i = blockIdx.x * blockDim.x + threadIdx.x;
    if (i < NN) deg[i] = 1.0f;  // self-loop
}

__global__ void k_accum_deg(const long long* __restrict__ col,
                            float* __restrict__ deg) {
    int e = blockIdx.x * blockDim.x + threadIdx.x;
    if (e < NE) atomicAdd(&deg[(int)col[e]], 1.0f);
}

__global__ void k_dinv(float* __restrict__ deg) {
    int i = blockIdx.x * blockDim.x + threadIdx.x;
    if (i < NN) deg[i] = rsqrtf(fmaxf(deg[i], 1e-12f));
}

// -------------------------------------------------- GEMM1: h0 = x @ W1 ------
// One wave per 16x16 tile of h0. K=256 -> 64 v_wmma_f32_16x16x4_f32 per tile.
// Block = 256 threads = 8 waves; wave w owns N-tile w (HID = 8*16).
__global__ __launch_bounds__(256) void k_gemm_xw1(const float* __restrict__ x,
                                                  const float* __restrict__ W1,
                                                  float* __restrict__ h0) {
    const int lane = threadIdx.x & 31;
    const int w    = threadIdx.x >> 5;   // N-tile index 0..7
    const int half = lane >> 4;          // K sub-pair selector
    const int m    = lane & 15;
    const int row  = blockIdx.x * 16 + m;    // A-matrix row (M)
    const int col  = w * 16 + m;             // B/D column (N)

    v8f c = {0.f,0.f,0.f,0.f,0.f,0.f,0.f,0.f};
    const float* __restrict__ arow = x + (size_t)row * INC + 2 * half;
    for (int k = 0; k < INC; k += 4) {
        v2f a, b;
        a[0] = arow[k];
        a[1] = arow[k + 1];
        b[0] = W1[(k + 2 * half)     * HID + col];
        b[1] = W1[(k + 2 * half + 1) * HID + col];
        c = __builtin_amdgcn_wmma_f32_16x16x4_f32(false, a, false, b,
                                                  (short)0, c, false, false);
    }
    const int r0 = blockIdx.x * 16 + 8 * half;
#pragma unroll
    for (int v = 0; v < 8; ++v)
        h0[(size_t)(r0 + v) * HID + col] = c[v];
}

// --------------------------------------------- normalized scatter (128 wide)
__global__ void k_self_init(const float* __restrict__ src,
                            float* __restrict__ dst,
                            const float* __restrict__ dinv) {
    int idx = blockIdx.x * blockDim.x + threadIdx.x;
    if (idx < NN * HID) {
        int i = idx >> 7;            // node
        float d = dinv[i];
        dst[idx] = src[idx] * d * d; // self-loop message
    }
}

__global__ void k_scatter(const float* __restrict__ src,
                          float* __restrict__ dst,
                          const long long* __restrict__ rows,
                          const long long* __restrict__ cols,
                          const float* __restrict__ dinv) {
    long long idx = (long long)blockIdx.x * blockDim.x + threadIdx.x;
    if (idx >= (long long)NE * (HID / 4)) return;
    int e = (int)(idx >> 5);            // HID/4 = 32 chunks per edge
    int f = ((int)idx & 31) * 4;
    int r = (int)rows[e];
    int t = (int)cols[e];
    float norm = dinv[r] * dinv[t];
    const float4 v = *(const float4*)(src + (size_t)r * HID + f);
    float* d = dst + (size_t)t * HID + f;
    atomicAdd(d + 0, v.x * norm);
    atomicAdd(d + 1, v.y * norm);
    atomicAdd(d + 2, v.z * norm);
    atomicAdd(d + 3, v.w * norm);
}

__global__ void k_relu_bias(float* __restrict__ h, const float* __restrict__ b1) {
    int idx = blockIdx.x * blockDim.x + threadIdx.x;
    if (idx < NN * HID) {
        int f = idx & (HID - 1);
        h[idx] = fmaxf(h[idx] + b1[f], 0.0f);
    }
}

// ------------------- GEMM2: mu = agg@Wmu + bmu ; lv = agg@Wlv + blv ---------
// Waves 0..3 -> mu columns, waves 4..7 -> logvar columns. K=128 -> 32 WMMAs.
__global__ __launch_bounds__(256) void k_gemm_out(const float* __restrict__ agg,
                                                  const float* __restrict__ Wmu,
                                                  const float* __restrict__ bmu,
                                                  const float* __restrict__ Wlv,
                                                  const float* __restrict__ blv,
                                                  float* __restrict__ out) {
    const int lane  = threadIdx.x & 31;
    const int w     = threadIdx.x >> 5;       // 0..7 (uniform per wave)
    const int half  = lane >> 4;
    const int m     = lane & 15;
    const bool islv = (w >= 4);
    const float* __restrict__ W    = islv ? Wlv : Wmu;
    const float* __restrict__ bias = islv ? blv : bmu;
    const int col = (islv ? (w - 4) : w) * 16 + m;  // 0..63
    const int row = blockIdx.x * 16 + m;

    v8f c = {0.f,0.f,0.f,0.f,0.f,0.f,0.f,0.f};
    const float* __restrict__ arow = agg + (size_t)row * HID + 2 * half;
    for (int k = 0; k < HID; k += 4) {
        v2f a, b;
        a[0] = arow[k];
        a[1] = arow[k + 1];
        b[0] = W[(k + 2 * half)     * LAT + col];
        b[1] = W[(k + 2 * half + 1) * LAT + col];
        c = __builtin_amdgcn_wmma_f32_16x16x4_f32(false, a, false, b,
                                                  (short)0, c, false, false);
    }
    float bb = bias[col];
    float* __restrict__ dst = out + (islv ? (size_t)NN * LAT : 0);
    const int r0 = blockIdx.x * 16 + 8 * half;
#pragma unroll
    for (int v = 0; v < 8; ++v)
        dst[(size_t)(r0 + v) * LAT + col] = c[v] + bb;
}

// ---------------------------------------------------------------------------
extern "C" void kernel_launch(void* const* d_in, const int* in_sizes, int n_in,
                              void* d_out, int out_size, void* d_ws, size_t ws_size,
                              hipStream_t stream) {
    const float*     x    = (const float*)d_in[0];
    const long long* ei   = (const long long*)d_in[1]; // [2, E] int64
    const float*     W1   = (const float*)d_in[2];
    const float*     b1   = (const float*)d_in[3];
    const float*     Wmu  = (const float*)d_in[4];
    const float*     bmu  = (const float*)d_in[5];
    const float*     Wlv  = (const float*)d_in[6];
    const float*     blv  = (const float*)d_in[7];
    float*           out  = (float*)d_out;

    const long long* rows = ei;        // edge_index[0] (source)
    const long long* cols = ei + NE;   // edge_index[1] (target)

    float* dinv = (float*)d_ws;                         // NN floats
    float* buf0 = dinv + NN;                            // NN*HID : h0, later agg2
    float* buf1 = buf0 + (size_t)NN * HID;              // NN*HID : agg1 -> h

    const int T = 256;
    const int gN    = (NN + T - 1) / T;
    const int gE    = (NE + T - 1) / T;
    const int gNH   = (NN * HID + T - 1) / T;
    const int gScat = (int)(((long long)NE * (HID / 4) + T - 1) / T);
    const int gTile = NN / 16;  // 3125, exact

    // degree -> dinv (stored in place)
    k_init_deg <<<gN, T, 0, stream>>>(dinv);
    k_accum_deg<<<gE, T, 0, stream>>>(cols, dinv);
    k_dinv     <<<gN, T, 0, stream>>>(dinv);

    // layer 1: h0 = x @ W1 ; agg1 = sym-norm scatter ; h = relu(agg1 + b1)
    k_gemm_xw1 <<<gTile, T, 0, stream>>>(x, W1, buf0);
    k_self_init<<<gNH,   T, 0, stream>>>(buf0, buf1, dinv);
    k_scatter  <<<gScat, T, 0, stream>>>(buf0, buf1, rows, cols, dinv);
    k_relu_bias<<<gNH,   T, 0, stream>>>(buf1, b1);

    // layer 2 aggregation first (linearity), then both output GEMMs fused
    k_self_init<<<gNH,   T, 0, stream>>>(buf1, buf0, dinv);
    k_scatter  <<<gScat, T, 0, stream>>>(buf1, buf0, rows, cols, dinv);
    k_gemm_out <<<gTile, T, 0, stream>>>(buf0, Wmu, bmu, Wlv, blv, out);
}